// Attention_48335561949593
// MI455X (gfx1250) — compile-verified
//
#include <hip/hip_runtime.h>
#include <math.h>

typedef __attribute__((ext_vector_type(2))) float v2f;
typedef __attribute__((ext_vector_type(8))) float v8f;

static constexpr int   Bb    = 8;
static constexpr int   Nn    = 4096;
static constexpr int   Kc    = 16;
static constexpr int   Dc    = 128;
static constexpr int   NPB   = 256;            // rows per block
static constexpr int   NBLK  = Nn / NPB;       // 16 blocks per batch
static constexpr int   ABP   = 132;            // LDS pitch for A/B coeff tiles
static constexpr int   GP    = 18;             // LDS pitch for gamma tile
static constexpr float EPSF  = 1e-08f;
static constexpr float TWOPI = 6.28318530717958647692f;

__device__ __forceinline__ v8f wmma4(v2f a, v2f b, v8f c) {
  // V_WMMA_F32_16X16X4_F32 : D(16x16 f32) = A(16x4 f32) * B(4x16 f32) + C
  return __builtin_amdgcn_wmma_f32_16x16x4_f32(
      /*neg_a=*/false, a, /*neg_b=*/false, b,
      /*c_mod=*/(short)0, c, /*reuse_a=*/false, /*reuse_b=*/false);
}

// DPP16 lane permute (stays within each 16-lane row -> halves independent).
template <int CTRL>
__device__ __forceinline__ float dpp_f(float v) {
  int i = __builtin_bit_cast(int, v);
  i = __builtin_amdgcn_mov_dpp(i, CTRL, 0xF, 0xF, true);
  return __builtin_bit_cast(float, i);
}

// Butterfly reductions over the 16 lanes of each row half (VALU-only, no LDS).
__device__ __forceinline__ float red16_max(float v) {
  v = fmaxf(v, dpp_f<0x0B1>(v));   // quad_perm(1,0,3,2)  : xor 1
  v = fmaxf(v, dpp_f<0x04E>(v));   // quad_perm(2,3,0,1)  : xor 2
  v = fmaxf(v, dpp_f<0x141>(v));   // ROW_HALF_MIRROR     : pairs 4-groups
  v = fmaxf(v, dpp_f<0x140>(v));   // ROW_MIRROR          : pairs 8-groups
  return v;
}
__device__ __forceinline__ float red16_sum(float v) {
  v += dpp_f<0x0B1>(v);
  v += dpp_f<0x04E>(v);
  v += dpp_f<0x141>(v);
  v += dpp_f<0x140>(v);
  return v;
}

// ---------------------------------------------------------------------------
// Main EM pass: logits GEMM -> softmax -> gamma moments GEMM (all WMMA f32).
// One block = 256 threads (8 waves) handling 256 rows of one batch.
// ---------------------------------------------------------------------------
__global__ __launch_bounds__(256) void gmm_em_pass(
    const float* __restrict__ x,      // [B][N][D]
    const float* __restrict__ Avec,   // [B][K][D]  mu/(sigma+eps)
    const float* __restrict__ Bvec,   // [B][K][D]  -0.5/(sigma+eps)
    const float* __restrict__ Cc,     // [B][K]     per-cluster constant
    float* __restrict__ S1part,       // [B][NBLK][K][D]
    float* __restrict__ S2part,       // [B][NBLK][K][D]
    float* __restrict__ Gpart)        // [B][NBLK][K]
{
  __shared__ float sA[Kc * ABP];
  __shared__ float sB[Kc * ABP];
  __shared__ float sC[Kc];
  __shared__ float sG[NPB * GP];
  __shared__ float sGs[8][Kc];

  const int b    = blockIdx.x / NBLK;
  const int nb   = blockIdx.x % NBLK;
  const int tid  = threadIdx.x;
  const int w    = tid >> 5;
  const int lane = tid & 31;
  const int half = lane >> 4;
  const int col  = lane & 15;

  const float* xb = x + ((size_t)b * Nn + (size_t)nb * NPB) * Dc;

  // Warm near caches for this block's X region (global_prefetch_b8).
  __builtin_prefetch(xb + (size_t)tid * Dc, 0, 0);

  // Stage per-cluster coefficient matrices into LDS (bank-padded).
  const float* Ag = Avec + (size_t)b * Kc * Dc;
  const float* Bg = Bvec + (size_t)b * Kc * Dc;
  for (int i = tid; i < Kc * Dc; i += 256) {
    int k = i >> 7, d = i & 127;
    sA[k * ABP + d] = Ag[i];
    sB[k * ABP + d] = Bg[i];
  }
  if (tid < Kc) sC[tid] = Cc[b * Kc + tid];
  __syncthreads();

  // ---------------- Phase 1: logits + softmax, 2 row-tiles per wave --------
  float gsum = 0.f;
  for (int tt = 0; tt < 2; ++tt) {
    const int tile = w * 2 + tt;
    const float* xrow = xb + (size_t)(tile * 16 + col) * Dc; // A-frag row
    v8f acc = {};
    #pragma unroll 8
    for (int d0 = 0; d0 < Dc; d0 += 4) {
      const int dd = d0 + 2 * half;
      v2f a;  a.x  = xrow[dd];       a.y  = xrow[dd + 1];
      v2f a2; a2.x = a.x * a.x;      a2.y = a.y * a.y;
      v2f bA; bA.x = sA[col * ABP + dd]; bA.y = sA[col * ABP + dd + 1];
      v2f bB; bB.x = sB[col * ABP + dd]; bB.y = sB[col * ABP + dd + 1];
      acc = wmma4(a,  bA, acc);   // + x  * (mu/sig)
      acc = wmma4(a2, bB, acc);   // + x^2 * (-0.5/sig)
    }
    // acc[r] = logits[row = r+8*half][k = col]; softmax over k (16 lanes/half)
    const float cK = sC[col];
    #pragma unroll
    for (int r = 0; r < 8; ++r) {
      const float v = acc[r] + cK;
      const float m = red16_max(v);
      const float e = __expf(v - m);
      const float s = red16_sum(e);
      const float g = e / s;
      sG[(tile * 16 + r + 8 * half) * GP + col] = g;
      gsum += g;
    }
  }
  // Per-k gamma sums: fold halves, then reduce waves via LDS (fixed order).
  gsum += __shfl_xor(gsum, 16, 32);
  if (half == 0) sGs[w][col] = gsum;
  __syncthreads();
  if (tid < Kc) {
    float t = 0.f;
    #pragma unroll
    for (int ww = 0; ww < 8; ++ww) t += sGs[ww][tid];
    Gpart[(b * NBLK + nb) * Kc + tid] = t;
  }

  // ---------------- Phase 2: S1 = gamma^T X, S2 = gamma^T X^2 --------------
  // Wave w owns d-chunk [w*16, w*16+16). Contraction over the block's rows.
  const int dbase = w * 16;
  v8f s1 = {};
  v8f s2 = {};
  for (int tile = 0; tile < NPB / 16; ++tile) {
    const float* xt = xb + (size_t)tile * 16 * Dc + dbase + col;
    #pragma unroll
    for (int t = 0; t < 4; ++t) {
      const int rA = 4 * t + 2 * half;
      v2f ga;
      ga.x = sG[(tile * 16 + rA)     * GP + col];
      ga.y = sG[(tile * 16 + rA + 1) * GP + col];
      v2f bx;
      bx.x = xt[(size_t)rA * Dc];
      bx.y = xt[(size_t)(rA + 1) * Dc];
      v2f bx2; bx2.x = bx.x * bx.x; bx2.y = bx.y * bx.y;
      s1 = wmma4(ga, bx,  s1);
      s2 = wmma4(ga, bx2, s2);
    }
  }
  // acc[r] -> [k = r+8*half][d = dbase+col]; per-block partials (no atomics).
  float* S1o = S1part + ((size_t)(b * NBLK + nb)) * Kc * Dc;
  float* S2o = S2part + ((size_t)(b * NBLK + nb)) * Kc * Dc;
  #pragma unroll
  for (int r = 0; r < 8; ++r) {
    const int k = r + 8 * half;
    S1o[(size_t)k * Dc + dbase + col] = s1[r];
    S2o[(size_t)k * Dc + dbase + col] = s2[r];
  }
}

// ---------------------------------------------------------------------------
// Setup / finalize: one block per (b,k), 128 threads over d.
// mode 0: init state from slots_mu / slots_log_sigma / mixing
// mode 1: reduce partials -> mu,sigma,pi -> next-iter Avec/Bvec/C
// mode 2: reduce partials -> slots = mu + sigma * noise  (output)
// ---------------------------------------------------------------------------
__global__ __launch_bounds__(128) void gmm_setup(
    int mode,
    const float* __restrict__ slots_mu,
    const float* __restrict__ slots_ls,
    const float* __restrict__ mix,
    const float* __restrict__ noise,
    const float* __restrict__ S1part,
    const float* __restrict__ S2part,
    const float* __restrict__ Gpart,
    float* __restrict__ Avec,
    float* __restrict__ Bvec,
    float* __restrict__ Cc,
    float* __restrict__ outp)
{
  __shared__ float red[Dc];
  const int b = blockIdx.x / Kc;
  const int k = blockIdx.x % Kc;
  const int d = threadIdx.x;

  float mu, sigma, pi;
  if (mode == 0) {
    mu    = slots_mu[k * Dc + d];
    sigma = expf(slots_ls[k * Dc + d]);
    pi    = mix[k];
  } else {
    float s1 = 0.f, s2 = 0.f, g = 0.f;
    for (int nb = 0; nb < NBLK; ++nb) {
      const size_t base = ((size_t)(b * NBLK + nb) * Kc + k) * Dc + d;
      s1 += S1part[base];
      s2 += S2part[base];
      g  += Gpart[(b * NBLK + nb) * Kc + k];
    }
    const float gs = g + EPSF;                 // gamma_sum + eps
    mu    = s1 / gs;
    sigma = (s2 - 2.f * mu * s1 + mu * mu * g) / gs;
    pi    = g / (float)Nn;                     // mean over N
  }

  if (mode == 2) {
    const size_t o = ((size_t)b * Kc + k) * Dc + d;
    outp[o] = mu + sigma * noise[o];
    return;
  }

  const float inv = 1.f / (sigma + EPSF);
  const size_t o = ((size_t)b * Kc + k) * Dc + d;
  Avec[o] = mu * inv;
  Bvec[o] = -0.5f * inv;

  red[d] = -0.5f * logf(TWOPI * sigma + EPSF) - 0.5f * mu * mu * inv;
  __syncthreads();
  for (int s = Dc / 2; s > 0; s >>= 1) {
    if (d < s) red[d] += red[d + s];
    __syncthreads();
  }
  if (d == 0) Cc[b * Kc + k] = red[0] + logf(pi + EPSF);
}

// ---------------------------------------------------------------------------
extern "C" void kernel_launch(void* const* d_in, const int* in_sizes, int n_in,
                              void* d_out, int out_size, void* d_ws, size_t ws_size,
                              hipStream_t stream) {
  const float* x     = (const float*)d_in[0];  // inputs        [B,N,D]
  const float* smu   = (const float*)d_in[1];  // slots_mu      [1,K,D]
  const float* sls   = (const float*)d_in[2];  // slots_log_sig [1,K,D]
  const float* mix   = (const float*)d_in[3];  // mixing        [1,K]
  const float* noise = (const float*)d_in[4];  // noise         [B,K,D]

  float* ws = (float*)d_ws;
  const size_t BKD  = (size_t)Bb * Kc * Dc;          // 16384
  const size_t PART = (size_t)Bb * NBLK * Kc * Dc;   // 262144
  float* Avec  = ws;
  float* Bvec  = Avec + BKD;
  float* Cc    = Bvec + BKD;                          // B*K
  float* Gpart = Cc + (size_t)Bb * Kc;                // B*NBLK*K
  float* S1p   = Gpart + (size_t)Bb * NBLK * Kc;
  float* S2p   = S1p + PART;

  dim3 gS(Bb * Kc), bS(Dc);
  dim3 gM(Bb * NBLK), bM(256);

  gmm_setup<<<gS, bS, 0, stream>>>(0, smu, sls, mix, nullptr,
                                   S1p, S2p, Gpart, Avec, Bvec, Cc, nullptr);
  for (int it = 0; it < 3; ++it) {
    gmm_em_pass<<<gM, bM, 0, stream>>>(x, Avec, Bvec, Cc, S1p, S2p, Gpart);
    gmm_setup<<<gS, bS, 0, stream>>>(it < 2 ? 1 : 2, smu, sls, mix, noise,
                                     S1p, S2p, Gpart, Avec, Bvec, Cc,
                                     (float*)d_out);
  }
}